// FarthestPointSampler_33698313404545
// MI455X (gfx1250) — compile-verified
//
#include <hip/hip_runtime.h>

// Farthest Point Sampling for MI455X (gfx1250).
// pos: [B=32, N=131072, C=3] f32 -> out: [B, 1024] int32 centroid indices.
//
// 1 workgroup per batch, 1024 threads = 32 wave32s (fills one WGP).
// dist[] lives in d_ws (L2-resident, 16 MiB). The first 12288 points of each
// batch (xyz AoS, 144 KiB) + their dist (48 KiB) are staged ONCE into LDS via
// CDNA5 async global->LDS b128 loads (ASYNCcnt / s_wait_asynccnt) and reused
// for all 1024 iterations. Argmax: wave32 shfl_xor tree + cross-wave LDS
// stage; the reduction carries the winning point's coords so the next
// centroid needs no dependent global fetch. 2 barriers per iteration.
//
// NOTE: the reference seeds the first centroid with a JAX threefry randint;
// not reproducible here, so we deterministically start at index 0.

#define FPS_INF 1e10f
#define THREADS 1024
#define LDS_PTS 12288                 // points staged in LDS per batch
#define KLDS    (LDS_PTS / THREADS)   // 12 points per thread in LDS region

#if defined(__has_builtin)
#  if __has_builtin(__builtin_amdgcn_global_load_async_to_lds_b128)
#    define FPS_ASYNC 1
#  endif
#endif
#ifndef FPS_ASYNC
#  define FPS_ASYNC 0
#endif

#if FPS_ASYNC
typedef int fps_v4i __attribute__((vector_size(16)));
typedef fps_v4i __attribute__((address_space(1))) fps_g4;   // global int4
typedef fps_v4i __attribute__((address_space(3))) fps_l4;   // LDS int4
#endif

__global__ void fps_init_dist(float* __restrict__ gDist, long long total) {
  long long i = (long long)blockIdx.x * blockDim.x + threadIdx.x;
  if (i < total) gDist[i] = FPS_INF;
}

__global__ __launch_bounds__(THREADS, 1)
void fps_main(const float* __restrict__ pos, float* __restrict__ gDist,
              int* __restrict__ out, int N, int NP) {
  const int b    = blockIdx.x;
  const int tid  = threadIdx.x;
  const int lane = tid & 31;
  const int w    = tid >> 5;
  const long long base = (long long)b * N;
  const float* gp = pos + base * 3;          // this batch's xyz stream (16B aligned)

  __shared__ __align__(16) float sP[3 * LDS_PTS];  // 144 KiB AoS xyz
  __shared__ float sD[LDS_PTS];                    // 48 KiB dist
  __shared__ float sRedV[32];
  __shared__ int   sRedI[32];
  __shared__ float sRedX[32], sRedY[32], sRedZ[32];
  __shared__ float sCX, sCY, sCZ;
  __shared__ int   sFar;

  // ---- one-time stage of the LDS-resident slice (9 x b128 per thread) ----
  #pragma unroll
  for (int k = 0; k < (3 * LDS_PTS) / (4 * THREADS); ++k) {
    int j = tid + k * THREADS;                 // float4 chunk index
#if FPS_ASYNC
    fps_g4* g = (fps_g4*)(gp + 4 * j);
    fps_l4* l = (fps_l4*)(&sP[4 * j]);
    __builtin_amdgcn_global_load_async_to_lds_b128(g, l, 0, 0);
#else
    *(float4*)(&sP[4 * j]) = *(const float4*)(gp + 4 * j);
#endif
  }
  #pragma unroll
  for (int k = 0; k < KLDS; ++k) sD[tid + k * THREADS] = FPS_INF;
#if FPS_ASYNC
  asm volatile("s_wait_asynccnt 0" ::: "memory");
#endif
  if (tid == 0) {                 // first centroid = point 0 (in LDS slice)
    sFar = 0;
    sCX = sP[0]; sCY = sP[1]; sCZ = sP[2];
  }
  __syncthreads();

  const int KMAX = N >> 10;       // 128 points per thread total

  for (int it = 0; it < NP; ++it) {
    const float cx = sCX, cy = sCY, cz = sCZ;
    if (tid == 0) out[(long long)b * NP + it] = sFar;

    float bv = -1.0f; int bi = 0;
    float bx = 0.f, by = 0.f, bz = 0.f;

    // ---- LDS-resident slice: zero global traffic ----
    #pragma unroll
    for (int k = 0; k < KLDS; ++k) {
      int j = tid + (k << 10);
      float x = sP[3 * j], y = sP[3 * j + 1], z = sP[3 * j + 2];
      float dx = x - cx, dy = y - cy, dz = z - cz;
      float d  = dx * dx + dy * dy + dz * dz;
      float old = sD[j];
      float nd  = fminf(old, d);
      if (nd < old) sD[j] = nd;
      if (nd > bv) { bv = nd; bi = j; bx = x; by = y; bz = z; }
    }
    // ---- L2-resident remainder: 12B pos + 4B dist read, rare dist write ----
    #pragma unroll 4
    for (int k = KLDS; k < KMAX; ++k) {
      int j = tid + (k << 10);
      const float* pp = gp + 3LL * j;
      float x = pp[0], y = pp[1], z = pp[2];
      float dx = x - cx, dy = y - cy, dz = z - cz;
      float d  = dx * dx + dy * dy + dz * dz;
      float old = gDist[base + j];
      float nd  = fminf(old, d);
      if (nd < old) gDist[base + j] = nd;
      if (nd > bv) { bv = nd; bi = j; bx = x; by = y; bz = z; }
    }

    // ---- wave32 argmax (value; ties -> lowest index, matching jnp.argmax) ----
    #pragma unroll
    for (int m = 16; m > 0; m >>= 1) {
      float ov = __shfl_xor(bv, m, 32);
      int   oi = __shfl_xor(bi, m, 32);
      float ox = __shfl_xor(bx, m, 32);
      float oy = __shfl_xor(by, m, 32);
      float oz = __shfl_xor(bz, m, 32);
      if (ov > bv || (ov == bv && oi < bi)) { bv = ov; bi = oi; bx = ox; by = oy; bz = oz; }
    }
    if (lane == 0) { sRedV[w] = bv; sRedI[w] = bi; sRedX[w] = bx; sRedY[w] = by; sRedZ[w] = bz; }
    __syncthreads();
    if (w == 0) {
      bv = sRedV[lane]; bi = sRedI[lane];
      bx = sRedX[lane]; by = sRedY[lane]; bz = sRedZ[lane];
      #pragma unroll
      for (int m = 16; m > 0; m >>= 1) {
        float ov = __shfl_xor(bv, m, 32);
        int   oi = __shfl_xor(bi, m, 32);
        float ox = __shfl_xor(bx, m, 32);
        float oy = __shfl_xor(by, m, 32);
        float oz = __shfl_xor(bz, m, 32);
        if (ov > bv || (ov == bv && oi < bi)) { bv = ov; bi = oi; bx = ox; by = oy; bz = oz; }
      }
      if (lane == 0) { sFar = bi; sCX = bx; sCY = by; sCZ = bz; }
    }
    __syncthreads();
  }
}

extern "C" void kernel_launch(void* const* d_in, const int* in_sizes, int n_in,
                              void* d_out, int out_size, void* d_ws, size_t ws_size,
                              hipStream_t stream) {
  const float* pos = (const float*)d_in[0];
  int* out = (int*)d_out;

  const int B = 32;                                   // fixed by reference setup
  const long long BN = (long long)in_sizes[0] / 3;    // B*N points
  const int N  = (int)(BN / B);                       // 131072
  const int NP = out_size / B;                        // 1024
  (void)n_in; (void)ws_size;

  float* gDist = (float*)d_ws;                        // BN*4 = 16 MiB scratch

  int initBlocks = (int)((BN + THREADS - 1) / THREADS);
  fps_init_dist<<<initBlocks, THREADS, 0, stream>>>(gDist, BN);
  fps_main<<<B, THREADS, 0, stream>>>(pos, gDist, out, N, NP);
}